// GraphFilter_65910568124751
// MI455X (gfx1250) — compile-verified
//
#include <hip/hip_runtime.h>
#include <hip/hip_bf16.h>

// ---------------------------------------------------------------------------
// GraphFilter: stable stream compaction of edges with d12 < CUTOFF^2,
// padded to OUT_SIZE with fill values. Two-pass deterministic compaction.
// Full-chunk input loads are issued as b128 clauses; gfx1250 prefetch hints used.
// ---------------------------------------------------------------------------

#define GF_BLOCK 256
#define GF_ITEMS 32
#define GF_CHUNK (GF_BLOCK * GF_ITEMS)   // 8192 elements per block

// ---------------------------- Pass 1: per-block counts ----------------------
__global__ __launch_bounds__(GF_BLOCK)
void gf_count_kernel(const float* __restrict__ d12, int n, float c2,
                     int* __restrict__ counts)
{
    const int bid  = blockIdx.x;
    const int tid  = threadIdx.x;
    const int base = bid * GF_CHUNK + tid * GF_ITEMS;

    // prefetch the next block's slice of d12 into L2 (global_prefetch_b8)
    if (base + GF_CHUNK < n)
        __builtin_prefetch(d12 + base + GF_CHUNK, 0, 0);

    int cnt = 0;
    if (base + GF_ITEMS <= n) {
        const float4* p = reinterpret_cast<const float4*>(d12 + base);
#pragma unroll
        for (int k = 0; k < GF_ITEMS / 4; ++k) {
            float4 v = p[k];
            cnt += (v.x < c2) + (v.y < c2) + (v.z < c2) + (v.w < c2);
        }
    } else {
        for (int k = 0; k < GF_ITEMS; ++k) {
            int i = base + k;
            if (i < n) cnt += (d12[i] < c2);
        }
    }

    __shared__ int lds[GF_BLOCK];
    lds[tid] = cnt;
    __syncthreads();
#pragma unroll
    for (int off = GF_BLOCK / 2; off > 0; off >>= 1) {
        if (tid < off) lds[tid] += lds[tid + off];
        __syncthreads();
    }
    if (tid == 0) counts[bid] = lds[0];
}

// ------------------- Pass 2: exclusive scan of block counts -----------------
// Single block of 1024 threads; handles any NB via chunked Hillis-Steele.
__global__ __launch_bounds__(1024)
void gf_scan_kernel(const int* __restrict__ counts,
                    int* __restrict__ offsets,
                    int* __restrict__ total_out, int nb)
{
    __shared__ int lds[1024];
    __shared__ int base;
    const int tid = threadIdx.x;
    if (tid == 0) base = 0;
    __syncthreads();

    for (int chunk = 0; chunk < nb; chunk += 1024) {
        const int i = chunk + tid;
        const int v = (i < nb) ? counts[i] : 0;
        lds[tid] = v;
        __syncthreads();
#pragma unroll
        for (int off = 1; off < 1024; off <<= 1) {
            int t = (tid >= off) ? lds[tid - off] : 0;
            __syncthreads();
            lds[tid] += t;
            __syncthreads();
        }
        const int incl = lds[tid];
        if (i < nb) offsets[i] = base + incl - v;   // global exclusive prefix
        const int chunk_total = lds[1023];
        __syncthreads();
        if (tid == 0) base += chunk_total;
        __syncthreads();
    }
    if (tid == 0) *total_out = base;
}

// ---------------------- Pass 3: compacting scatter --------------------------
// Bulk-loads d12 / esrc / edst chunks as b128 clauses; the esrc/edst loads are
// issued before the block scan so they overlap the barrier latency.
__global__ __launch_bounds__(GF_BLOCK)
void gf_scatter_kernel(const float* __restrict__ d12,
                       const int* __restrict__ esrc,
                       const int* __restrict__ edst,
                       int n, float c2,
                       const int* __restrict__ offsets,
                       int* __restrict__ out_src,
                       int* __restrict__ out_dst,
                       float* __restrict__ out_d12,
                       int* __restrict__ out_idx,
                       int out_cap)
{
    const int bid  = blockIdx.x;
    const int tid  = threadIdx.x;
    const int base = bid * GF_CHUNK + tid * GF_ITEMS;
    const bool full = (base + GF_ITEMS <= n);

    if (base + GF_CHUNK < n) {
        __builtin_prefetch(d12  + base + GF_CHUNK, 0, 0);
        __builtin_prefetch(esrc + base + GF_CHUNK, 0, 0);
        __builtin_prefetch(edst + base + GF_CHUNK, 0, 0);
    }

    float f[GF_ITEMS];
    int   sv[GF_ITEMS];
    int   dv[GF_ITEMS];
    int cnt = 0;
    if (full) {
        const float4* pf = reinterpret_cast<const float4*>(d12 + base);
#pragma unroll
        for (int k = 0; k < GF_ITEMS / 4; ++k) {
            float4 v = pf[k];
            f[4 * k + 0] = v.x; f[4 * k + 1] = v.y;
            f[4 * k + 2] = v.z; f[4 * k + 3] = v.w;
        }
#pragma unroll
        for (int k = 0; k < GF_ITEMS; ++k) cnt += (f[k] < c2);
        // bulk-load the index payloads; overlaps with the LDS scan below
        const int4* ps = reinterpret_cast<const int4*>(esrc + base);
        const int4* pd = reinterpret_cast<const int4*>(edst + base);
#pragma unroll
        for (int k = 0; k < GF_ITEMS / 4; ++k) {
            int4 s = ps[k];
            sv[4 * k + 0] = s.x; sv[4 * k + 1] = s.y;
            sv[4 * k + 2] = s.z; sv[4 * k + 3] = s.w;
            int4 d = pd[k];
            dv[4 * k + 0] = d.x; dv[4 * k + 1] = d.y;
            dv[4 * k + 2] = d.z; dv[4 * k + 3] = d.w;
        }
    } else {
#pragma unroll
        for (int k = 0; k < GF_ITEMS; ++k) {
            int i = base + k;
            bool ok = (i < n);
            f[k]  = ok ? d12[i]  : c2;   // >= c2 => fails predicate
            sv[k] = ok ? esrc[i] : 0;
            dv[k] = ok ? edst[i] : 0;
            cnt += (f[k] < c2);
        }
    }

    // intra-block exclusive scan of per-thread counts (thread chunks are
    // contiguous and ordered, so global output order == input index order)
    __shared__ int lds[GF_BLOCK];
    lds[tid] = cnt;
    __syncthreads();
#pragma unroll
    for (int off = 1; off < GF_BLOCK; off <<= 1) {
        int t = (tid >= off) ? lds[tid - off] : 0;
        __syncthreads();
        lds[tid] += t;
        __syncthreads();
    }
    const int excl = lds[tid] - cnt;

    int pos = offsets[bid] + excl;
#pragma unroll
    for (int k = 0; k < GF_ITEMS; ++k) {
        if (f[k] < c2) {
            if (pos < out_cap) {
                out_idx[pos] = base + k;
                out_d12[pos] = f[k];
                out_src[pos] = sv[k];
                out_dst[pos] = dv[k];
            }
            ++pos;
        }
    }
}

// ---------------------- Pass 4: pad-fill the tail ---------------------------
__global__ __launch_bounds__(GF_BLOCK)
void gf_fill_kernel(int* __restrict__ out_src,
                    int* __restrict__ out_dst,
                    float* __restrict__ out_d12,
                    int* __restrict__ out_idx,
                    const int* __restrict__ total_p,
                    int out_cap, int fill_sd, int fill_idx, float fill_d)
{
    const int pos = blockIdx.x * GF_BLOCK + threadIdx.x;
    if (pos >= out_cap) return;
    int total = *total_p;
    if (total > out_cap) total = out_cap;
    if (pos >= total) {
        out_src[pos] = fill_sd;
        out_dst[pos] = fill_sd;
        out_d12[pos] = fill_d;
        out_idx[pos] = fill_idx;
    }
}

// ---------------------------------------------------------------------------
extern "C" void kernel_launch(void* const* d_in, const int* in_sizes, int n_in_arrs,
                              void* d_out, int out_size, void* d_ws, size_t ws_size,
                              hipStream_t stream)
{
    (void)n_in_arrs; (void)ws_size;

    const float* d12  = reinterpret_cast<const float*>(d_in[0]);
    const int*   esrc = reinterpret_cast<const int*>(d_in[1]);
    const int*   edst = reinterpret_cast<const int*>(d_in[2]);

    const int n      = in_sizes[0];      // N_EDGES_IN
    const int nattot = in_sizes[3];      // N_ATOMS (species length)
    const int out_n  = out_size / 4;     // OUT_SIZE per returned array
    const float c2   = 25.0f;            // CUTOFF^2

    // output layout: src | dst | d12 | indices
    int*   out_src = reinterpret_cast<int*>(d_out);
    int*   out_dst = out_src + out_n;
    float* out_d12 = reinterpret_cast<float*>(d_out) + 2 * (size_t)out_n;
    int*   out_idx = out_src + 3 * (size_t)out_n;

    const int nb = (n + GF_CHUNK - 1) / GF_CHUNK;

    int* ws      = reinterpret_cast<int*>(d_ws);
    int* counts  = ws;            // [nb]
    int* offsets = ws + nb;       // [nb]
    int* total_p = ws + 2 * nb;   // [1]

    gf_count_kernel<<<nb, GF_BLOCK, 0, stream>>>(d12, n, c2, counts);
    gf_scan_kernel<<<1, 1024, 0, stream>>>(counts, offsets, total_p, nb);
    gf_scatter_kernel<<<nb, GF_BLOCK, 0, stream>>>(d12, esrc, edst, n, c2, offsets,
                                                   out_src, out_dst, out_d12, out_idx,
                                                   out_n);
    gf_fill_kernel<<<(out_n + GF_BLOCK - 1) / GF_BLOCK, GF_BLOCK, 0, stream>>>(
        out_src, out_dst, out_d12, out_idx, total_p, out_n, nattot, n, c2);
}